// WarpKeyframe_27350351741414
// MI455X (gfx1250) — compile-verified
//
#include <hip/hip_runtime.h>
#include <hip/hip_bf16.h>

// Problem constants (fixed by setup_inputs in the reference)
#define B_   4
#define C_   64
#define T_   8
#define H_   128
#define W_   256

// One thread per output pixel (b,t,y,x); loops over the 64 channels.
// Bilinear weights/indices are computed once and amortized across channels.
// Keyframe gathers use default (RT) caching so the 33.5MB keyframe stays
// L2-resident; flow loads and the 268MB output stream use non-temporal hints.
__global__ __launch_bounds__(256) void WarpKeyframe_kernel(
    const float* __restrict__ kf,    // (B, C, 1, H, W)
    const float* __restrict__ flow,  // (B, 2, T, H, W)
    float* __restrict__ out)         // (B, C, T, H, W)
{
    constexpr int HW  = H_ * W_;
    constexpr int THW = T_ * HW;

    const int id = blockIdx.x * 256 + threadIdx.x;       // pixel id in [0, B*T*H*W)
    const int x  = id & (W_ - 1);                         // W = 256
    const int y  = (id >> 8) & (H_ - 1);                  // H = 128
    const int bt = id >> 15;                              // / (H*W)
    const int t  = bt & (T_ - 1);                         // T = 8
    const int b  = bt >> 3;

    // ---- flow load (streamed once; non-temporal) ----
    const int fbase = ((b * 2) * T_ + t) * HW + y * W_ + x;
    const float fx = __builtin_nontemporal_load(flow + fbase);            // channel 0 (x)
    const float fy = __builtin_nontemporal_load(flow + fbase + T_ * HW);  // channel 1 (y)

    // ---- sample coordinates: normalize/denormalize cancels algebraically ----
    const float ix = fminf(fmaxf((float)x + fx, 0.0f), (float)(W_ - 1));
    const float iy = fminf(fmaxf((float)y + fy, 0.0f), (float)(H_ - 1));
    const float x0f = floorf(ix);
    const float y0f = floorf(iy);
    const float wx = ix - x0f;
    const float wy = iy - y0f;
    const int x0 = (int)x0f;                 // in [0, W-1] by construction
    const int y0 = (int)y0f;
    const int x1 = min(x0 + 1, W_ - 1);
    const int y1 = min(y0 + 1, H_ - 1);

    const float w00 = (1.0f - wx) * (1.0f - wy);
    const float w01 = wx * (1.0f - wy);
    const float w10 = (1.0f - wx) * wy;
    const float w11 = wx * wy;

    const int i00 = y0 * W_ + x0;
    const int i01 = y0 * W_ + x1;
    const int i10 = y1 * W_ + x0;
    const int i11 = y1 * W_ + x1;

    const float* __restrict__ kb = kf + (size_t)b * C_ * HW;
    float* __restrict__ ob = out + (size_t)((b * C_) * T_ + t) * HW + y * W_ + x;

    // ---- channel loop: 4 gathers + 1 NT store per channel, chunked by 4 ----
    for (int c = 0; c < C_; c += 4) {
        const float* __restrict__ p = kb + (size_t)c * HW;

        // Warm WGP$/L2 for the next 128KB-strided chunk (gfx1250 global_prefetch_b8).
        if (c + 4 < C_) {
            __builtin_prefetch(p + 4 * HW + i00, 0, 3);
            __builtin_prefetch(p + 4 * HW + i11, 0, 3);
        }

        #pragma unroll
        for (int u = 0; u < 4; ++u) {
            const float* __restrict__ q = p + (size_t)u * HW;
            const float v00 = q[i00];
            const float v01 = q[i01];
            const float v10 = q[i10];
            const float v11 = q[i11];
            const float v = fmaf(w00, v00, fmaf(w01, v01, fmaf(w10, v10, w11 * v11)));
            __builtin_nontemporal_store(v, ob + (size_t)(c + u) * THW);
        }
    }
}

extern "C" void kernel_launch(void* const* d_in, const int* in_sizes, int n_in,
                              void* d_out, int out_size, void* d_ws, size_t ws_size,
                              hipStream_t stream) {
    const float* kf   = (const float*)d_in[0];  // (4, 64, 1, 128, 256) f32
    const float* flow = (const float*)d_in[1];  // (4, 2, 8, 128, 256) f32
    float* out = (float*)d_out;                 // (4, 64, 8, 128, 256) f32

    constexpr int total_pixels = B_ * T_ * H_ * W_;  // 1,048,576
    constexpr int block = 256;                        // 8 wave32 waves
    constexpr int grid = total_pixels / block;        // 4096 blocks

    WarpKeyframe_kernel<<<grid, block, 0, stream>>>(kf, flow, out);
}